// QuantumConv2d_9354438771406
// MI455X (gfx1250) — compile-verified
//
#include <hip/hip_runtime.h>
#include <hip/hip_bf16.h>

typedef float v2f __attribute__((ext_vector_type(2)));
typedef float v8f __attribute__((ext_vector_type(8)));

// ---------------------------------------------------------------------------
// Prep: build A = 0.25 * CNOT_ring(4) @ RX(weight) as separate real/imag
// 16x16 f32 matrices in workspace.
//   RX[j,k] = (-i)^m * prod_q (d_q ? sin(w_q/2) : cos(w_q/2)),  d = j^k, m=popc(d)
//   CNOT ring permutes rows: b1^=b0; b2^=b1; b3^=b2; b0^=b3  (qubit q -> bit q-1)
// ---------------------------------------------------------------------------
__global__ void qc_prep_kernel(const float* __restrict__ weight,
                               float* __restrict__ Ar, float* __restrict__ Ai) {
    int t = threadIdx.x;            // 0..255
    int j = t >> 4;                 // RX source row
    int k = t & 15;                 // column
    int d = j ^ k;
    float p = 0.25f;                // fold the Hadamard 1/4 scale into A
#pragma unroll
    for (int q = 0; q < 4; ++q) {
        float hw = 0.5f * weight[q];
        p *= ((d >> q) & 1) ? sinf(hw) : cosf(hw);
    }
    int m = __popc(d) & 3;          // (-i)^m
    float re = 0.0f, im = 0.0f;
    if (m == 0)      re =  p;
    else if (m == 1) im = -p;
    else if (m == 2) re = -p;
    else             im =  p;
    // CNOT-ring row permutation f(j)
    int b0 = j & 1, b1 = (j >> 1) & 1, b2 = (j >> 2) & 1, b3 = (j >> 3) & 1;
    b1 ^= b0; b2 ^= b1; b3 ^= b2; b0 ^= b3;
    int fj = b0 | (b1 << 1) | (b2 << 2) | (b3 << 3);
    Ar[fj * 16 + k] = re;
    Ai[fj * 16 + k] = im;
}

// ---------------------------------------------------------------------------
// FWHT butterfly stage across lanes (columns) via ds_swizzle XOR swap.
// s = +1 if (lane & D)==0 else -1;  v <- swz(v, xor D) + s*v
// ---------------------------------------------------------------------------
template <int OFF>
__device__ __forceinline__ void fwht_stage(v8f& v, float s) {
#pragma unroll
    for (int g = 0; g < 8; ++g) {
        float sw = __int_as_float(
            __builtin_amdgcn_ds_swizzle(__float_as_int(v[g]), OFF));
        v[g] = fmaf(s, v[g], sw);
    }
}

// ---------------------------------------------------------------------------
// Main: one wave32 per group of 8 batch elements.
//   T = A @ X       via v_wmma_f32_16x16x4_f32 (4 k-chunks, complex -> 16 WMMA)
//   Y = T @ H       via in-register Walsh-Hadamard transform over columns
// Output: interleaved complex64 (float2) stores.
// ---------------------------------------------------------------------------
__global__ void __launch_bounds__(256)
qc_main_kernel(const float* __restrict__ xr, const float* __restrict__ xi,
               const float* __restrict__ Ar, const float* __restrict__ Ai,
               float* __restrict__ out, int nElems) {
    const int lane = threadIdx.x & 31;
    const int wave = blockIdx.x * (blockDim.x >> 5) + (threadIdx.x >> 5);
    const int c = lane & 15;   // column / M index
    const int h = lane >> 4;   // lane half

    // A-operand fragments (16x4 f32 layout): vgpr r of chunk kc, lane l holds
    // A[M = l&15][K = 4*kc + 2*(l>>4) + r]
    v2f ar[4], ai[4], nai[4];
#pragma unroll
    for (int kc = 0; kc < 4; ++kc) {
#pragma unroll
        for (int r = 0; r < 2; ++r) {
            int K = 4 * kc + 2 * h + r;
            float vr = Ar[c * 16 + K];
            float vi = Ai[c * 16 + K];
            ar[kc][r]  = vr;
            ai[kc][r]  = vi;
            nai[kc][r] = -vi;
        }
    }

    // FWHT per-lane signs for stages d = 1,2,4,8
    const float s1 = (lane & 1) ? -1.0f : 1.0f;
    const float s2 = (lane & 2) ? -1.0f : 1.0f;
    const float s4 = (lane & 4) ? -1.0f : 1.0f;
    const float s8 = (lane & 8) ? -1.0f : 1.0f;

    const long base = (long)wave * 8;
    for (int e = 0; e < 8; ++e) {
        long elem = base + e;
        if (elem >= nElems) break;              // wave-uniform: EXEC stays full

        const float* xrb = xr + elem * 256;
        const float* xib = xi + elem * 256;

        // B-operand fragments (4x16 f32): vgpr r of chunk kc, lane l holds
        // X[K = 4*kc + 2*(l>>4) + r][N = l&15]
        v2f bxr[4], bxi[4];
#pragma unroll
        for (int kc = 0; kc < 4; ++kc) {
#pragma unroll
            for (int r = 0; r < 2; ++r) {
                int row = 4 * kc + 2 * h + r;
                bxr[kc][r] = xrb[row * 16 + c];
                bxi[kc][r] = xib[row * 16 + c];
            }
        }

        v8f tr = {0.f, 0.f, 0.f, 0.f, 0.f, 0.f, 0.f, 0.f};
        v8f ti = {0.f, 0.f, 0.f, 0.f, 0.f, 0.f, 0.f, 0.f};
#pragma unroll
        for (int kc = 0; kc < 4; ++kc) {
            // Tr = Ar*Xr - Ai*Xi ; Ti = Ar*Xi + Ai*Xr
            tr = __builtin_amdgcn_wmma_f32_16x16x4_f32(false, ar[kc],  false, bxr[kc], (short)0, tr, false, false);
            tr = __builtin_amdgcn_wmma_f32_16x16x4_f32(false, nai[kc], false, bxi[kc], (short)0, tr, false, false);
            ti = __builtin_amdgcn_wmma_f32_16x16x4_f32(false, ar[kc],  false, bxi[kc], (short)0, ti, false, false);
            ti = __builtin_amdgcn_wmma_f32_16x16x4_f32(false, ai[kc],  false, bxr[kc], (short)0, ti, false, false);
        }

        // Y = T @ H4 : Walsh-Hadamard transform along columns (lanes).
        // ds_swizzle group-of-32 offset: {xor_mask<<10 | or<<5 | and_mask(0x1f)}
        fwht_stage<0x041f>(tr, s1); fwht_stage<0x041f>(ti, s1);   // xor 1
        fwht_stage<0x081f>(tr, s2); fwht_stage<0x081f>(ti, s2);   // xor 2
        fwht_stage<0x101f>(tr, s4); fwht_stage<0x101f>(ti, s4);   // xor 4
        fwht_stage<0x201f>(tr, s8); fwht_stage<0x201f>(ti, s8);   // xor 8

        // C/D layout: vgpr g, lane l holds Y[M = g + 8*(l>>4)][N = l&15].
        // Store interleaved complex64.
        float2* ob = reinterpret_cast<float2*>(out) + elem * 256;
#pragma unroll
        for (int g = 0; g < 8; ++g) {
            int row = g + 8 * h;
            ob[row * 16 + c] = make_float2(tr[g], ti[g]);
        }
    }
}

extern "C" void kernel_launch(void* const* d_in, const int* in_sizes, int n_in,
                              void* d_out, int out_size, void* d_ws, size_t ws_size,
                              hipStream_t stream) {
    const float* weight = (const float*)d_in[0];
    const float* x_real = (const float*)d_in[1];
    const float* x_imag = (const float*)d_in[2];

    float* Ar = (float*)d_ws;        // 256 floats
    float* Ai = Ar + 256;            // 256 floats

    qc_prep_kernel<<<1, 256, 0, stream>>>(weight, Ar, Ai);

    const int B = in_sizes[1] / 256;             // 131072 batch elements
    const int wavesNeeded = (B + 7) / 8;         // 8 elements per wave
    const int blocks = (wavesNeeded + 7) / 8;    // 8 waves per 256-thread block

    qc_main_kernel<<<blocks, 256, 0, stream>>>(x_real, x_imag, Ar, Ai,
                                               (float*)d_out, B);
}